// DigitCaps_39522289057820
// MI455X (gfx1250) — compile-verified
//
#include <hip/hip_runtime.h>

// ---------------------------------------------------------------------------
// DigitCaps dynamic routing, fully fused for MI455X (gfx1250, wave32, WMMA).
//
// u_hat is NEVER materialized (would be 188.7MB, ~5 HBM passes in the naive
// schedule). Routing logits are linear in v, so we track Vsum = sum of v's
// and recompute u_hat tiles with V_WMMA_F32_16X16X4_F32 each iteration.
// ---------------------------------------------------------------------------

typedef __attribute__((ext_vector_type(2))) float v2f;
typedef __attribute__((ext_vector_type(8))) float v8f;

#define B_TOT    256
#define IN_CAPS  1152
#define IN_DIM   8
#define NUM_CAPS 10
#define DIM      16
#define EPSV     1e-7f

#define N_CHUNKS   36           // 1152 / 32
#define I_PER_CHUNK 32
#define WAVES_PER_WG 4
#define S_STRIDE  (B_TOT * NUM_CAPS * DIM)   // 40960 floats per chunk-partial

// ---------------------------------------------------------------------------
__global__ void caps_init(float* __restrict__ vsum) {
    int idx = blockIdx.x * blockDim.x + threadIdx.x;
    if (idx < S_STRIDE) vsum[idx] = 0.0f;
}

// ---------------------------------------------------------------------------
// One routing pass: for every (b-tile, i) recompute u_hat tile via WMMA f32,
// logits = u_hat . Vsum (butterfly-reduce over d), softmax over j,
// s-partial += c * u_hat. Deterministic: each WG writes its own ws region.
// ---------------------------------------------------------------------------
__launch_bounds__(WAVES_PER_WG * 32)
__global__ void caps_route(const float* __restrict__ x,
                           const float* __restrict__ W,
                           const float* __restrict__ vsum,
                           float* __restrict__ s_part) {
    const int lane  = threadIdx.x & 31;
    const int wave  = threadIdx.x >> 5;
    const int chunk = blockIdx.x;          // 0..35  (i range)
    const int btile = blockIdx.y;          // 0..15  (16 batches each)
    const int b0    = btile * 16;
    const int hi    = lane >> 4;           // half-wave select
    const int lm    = lane & 15;           // A-matrix row m / B,C column (d)

    // Vsum tile in WMMA C/D layout: vs[j][v] = Vsum[b0+v+8*hi, j, d=lm]
    v8f vs[NUM_CAPS];
    #pragma unroll
    for (int j = 0; j < NUM_CAPS; ++j) {
        #pragma unroll
        for (int v = 0; v < 8; ++v)
            vs[j][v] = vsum[((b0 + v + 8 * hi) * NUM_CAPS + j) * DIM + lm];
    }

    v8f s_acc[NUM_CAPS] = {};

    for (int ii = wave; ii < I_PER_CHUNK; ii += WAVES_PER_WG) {
        const int i = chunk * I_PER_CHUNK + ii;

        // ---- u_hat tile: A = x[b0:b0+16, i, 0:8], B = W[i] (8 x 160) ------
        v8f u[NUM_CAPS] = {};
        #pragma unroll
        for (int ks = 0; ks < 2; ++ks) {
            // A 16x4 f32 layout: lane row m=lm, VGPR0 K=2*hi, VGPR1 K=2*hi+1
            const float* xp = x + ((b0 + lm) * IN_CAPS + i) * IN_DIM
                                + 4 * ks + 2 * hi;
            v2f a; a.x = xp[0]; a.y = xp[1];
            #pragma unroll
            for (int j = 0; j < NUM_CAPS; ++j) {
                // B 4x16 f32 layout: lane col d=lm, VGPR0 K=2*hi, VGPR1 K=2*hi+1
                const float* wp = W + (((i * NUM_CAPS + j) * IN_DIM
                                        + 4 * ks + 2 * hi) * DIM + lm);
                v2f bm; bm.x = wp[0]; bm.y = wp[DIM];
                u[j] = __builtin_amdgcn_wmma_f32_16x16x4_f32(
                           false, a, false, bm, (short)0, u[j], false, false);
            }
        }

        // ---- logits[m,j] = sum_d u_hat[m,j,d] * Vsum[m,j,d] ---------------
        float lg[NUM_CAPS][8];
        #pragma unroll
        for (int j = 0; j < NUM_CAPS; ++j)
            #pragma unroll
            for (int v = 0; v < 8; ++v)
                lg[j][v] = u[j][v] * vs[j][v];
        // reduce over d: xor masks 1,2,4,8 stay inside each 16-lane half
        #pragma unroll
        for (int m = 1; m <= 8; m <<= 1)
            #pragma unroll
            for (int j = 0; j < NUM_CAPS; ++j)
                #pragma unroll
                for (int v = 0; v < 8; ++v)
                    lg[j][v] += __shfl_xor(lg[j][v], m, 32);

        // ---- softmax over j (redundant across lanes of a half; cheap) -----
        #pragma unroll
        for (int v = 0; v < 8; ++v) {
            float mx = lg[0][v];
            #pragma unroll
            for (int j = 1; j < NUM_CAPS; ++j) mx = fmaxf(mx, lg[j][v]);
            float sum = 0.0f;
            #pragma unroll
            for (int j = 0; j < NUM_CAPS; ++j) {
                float e = __expf(lg[j][v] - mx);
                lg[j][v] = e;
                sum += e;
            }
            float inv = 1.0f / sum;
            #pragma unroll
            for (int j = 0; j < NUM_CAPS; ++j) lg[j][v] *= inv;
        }

        // ---- s += c * u_hat (stays in accumulator layout) -----------------
        #pragma unroll
        for (int j = 0; j < NUM_CAPS; ++j)
            #pragma unroll
            for (int v = 0; v < 8; ++v)
                s_acc[j][v] = fmaf(lg[j][v], u[j][v], s_acc[j][v]);
    }

    // ---- combine the 4 waves in LDS, write chunk-partial (no atomics) -----
    __shared__ float sl[WAVES_PER_WG][NUM_CAPS][8][32];
    #pragma unroll
    for (int j = 0; j < NUM_CAPS; ++j)
        #pragma unroll
        for (int v = 0; v < 8; ++v)
            sl[wave][j][v][lane] = s_acc[j][v];
    __syncthreads();

    float* dst = s_part + (size_t)chunk * S_STRIDE;
    for (int t = threadIdx.x; t < NUM_CAPS * 8 * 32; t += WAVES_PER_WG * 32) {
        int j = t >> 8;
        int v = (t >> 5) & 7;
        int l = t & 31;
        float val = sl[0][j][v][l] + sl[1][j][v][l]
                  + sl[2][j][v][l] + sl[3][j][v][l];
        int b = b0 + v + 8 * (l >> 4);
        int d = l & 15;
        dst[(b * NUM_CAPS + j) * DIM + d] = val;
    }
}

// ---------------------------------------------------------------------------
// Reduce chunk partials (fixed order -> deterministic), squash, update Vsum,
// write output on the last routing iteration.
// grid = 256 (b), block = 160 (j*16+d): each 16-lane half of a wave = one j.
// ---------------------------------------------------------------------------
__global__ void caps_squash(const float* __restrict__ s_part,
                            float* __restrict__ vsum,
                            float* __restrict__ out,
                            int write_out) {
    const int b = blockIdx.x;
    const int t = threadIdx.x;          // 0..159
    const int idx = b * NUM_CAPS * DIM + t;

    float s = 0.0f;
    #pragma unroll 4
    for (int c = 0; c < N_CHUNKS; ++c)
        s += s_part[(size_t)c * S_STRIDE + idx];

    float sq = s * s;
    #pragma unroll
    for (int m = 1; m <= 8; m <<= 1)      // reduce over d within 16-lane half
        sq += __shfl_xor(sq, m, 32);

    float scale = sq / (1.0f + sq) * rsqrtf(sq + EPSV);
    float v = scale * s;

    vsum[idx] += v;
    if (write_out) out[idx] = v;
}

// ---------------------------------------------------------------------------
extern "C" void kernel_launch(void* const* d_in, const int* in_sizes, int n_in,
                              void* d_out, int out_size, void* d_ws, size_t ws_size,
                              hipStream_t stream) {
    const float* x = (const float*)d_in[0];   // [256,1152,8]
    const float* W = (const float*)d_in[1];   // [1152,10,8,16]
    float* out = (float*)d_out;               // [256,10,16]

    float* s_part = (float*)d_ws;                       // 36 * 40960 floats
    float* vsum   = s_part + (size_t)N_CHUNKS * S_STRIDE; // 40960 floats

    caps_init<<<(S_STRIDE + 255) / 256, 256, 0, stream>>>(vsum);

    for (int it = 0; it < 3; ++it) {
        caps_route<<<dim3(N_CHUNKS, 16), WAVES_PER_WG * 32, 0, stream>>>(
            x, W, vsum, s_part);
        caps_squash<<<B_TOT, NUM_CAPS * DIM, 0, stream>>>(
            s_part, vsum, out, it == 2 ? 1 : 0);
    }
}